// sparseGO_nn_60275571032433
// MI455X (gfx1250) — compile-verified
//
#include <hip/hip_runtime.h>
#include <hip/hip_bf16.h>

#define BATCH 256
#define XLD   8048      // x row stride (6000 gene + 2048 drug)
#define GENE  6000

typedef __attribute__((ext_vector_type(2))) float v2f;
typedef __attribute__((ext_vector_type(8))) float v8f;

// ---------------------------------------------------------------------------
// Zero a float region (all scatter/GEMM accumulators must start at 0).
__global__ void zero_f(float* __restrict__ p, int n) {
    int i = blockIdx.x * blockDim.x + threadIdx.x;
    if (i < n) p[i] = 0.f;
}

// ---------------------------------------------------------------------------
// Transpose gene slice of x: x[b, c] (row-major, stride XLD) -> xT[c, b].
__global__ void transpose_gene(const float* __restrict__ x, float* __restrict__ xT) {
    const int c = blockIdx.x;    // 0..GENE-1
    const int b = threadIdx.x;   // 0..255
    xT[c * BATCH + b] = x[b * XLD + c];
}

// ---------------------------------------------------------------------------
// Sparse linear scatter: out[row, b] += xT[col, b] * w[e].
// One block per edge, 256 lanes = batch. rows/cols/w are block-uniform
// (scalar loads); atomics are lane-coalesced (256 consecutive floats).
__global__ void sparse_scatter(const float* __restrict__ xT,
                               const int*   __restrict__ rows,
                               const int*   __restrict__ cols,
                               const float* __restrict__ w,
                               float*       __restrict__ out) {
    const int e = blockIdx.x;
    const int b = threadIdx.x;
    const int r = rows[e];
    const int c = cols[e];
    const float wv = w[e];
    atomicAdd(&out[r * BATCH + b], xT[c * BATCH + b] * wv);
}

// ---------------------------------------------------------------------------
// Fused bias + tanh + batch-norm over batch axis.
// pre/out in [nfeat, BATCH] layout; one block (256 threads) per feature.
__global__ void bias_tanh_bn(const float* __restrict__ pre,
                             const float* __restrict__ bias,
                             const float* __restrict__ gam,
                             const float* __restrict__ bet,
                             float*       __restrict__ out) {
    __shared__ float s1[BATCH];
    __shared__ float s2[BATCH];
    const int f = blockIdx.x;
    const int t = threadIdx.x;
    const float h = tanhf(pre[f * BATCH + t] + bias[f]);
    s1[t] = h;
    s2[t] = h * h;
    __syncthreads();
    for (int s = BATCH / 2; s > 0; s >>= 1) {
        if (t < s) { s1[t] += s1[t + s]; s2[t] += s2[t + s]; }
        __syncthreads();
    }
    const float m = s1[0] * (1.0f / BATCH);
    const float v = s2[0] * (1.0f / BATCH) - m * m;
    out[f * BATCH + t] = gam[f] * (h - m) * rsqrtf(v + 1e-5f) + bet[f];
}

// ---------------------------------------------------------------------------
// f32 WMMA GEMM with K-split: out[n, m] += sum_{k in chunk} A[m,k] * W[n,k]
//   (M = BATCH = 256; out pre-zeroed; blockIdx.z selects K chunk)
//   A_FEATMAJOR=false : A[m,k] = A[m*lda + k]  (batch-major, e.g. drug slice)
//   A_FEATMAJOR=true  : A[m,k] = A[k*lda + m]  (feature-major intermediates)
// One wave per 16x16 output tile; V_WMMA_F32_16X16X4_F32 per 4 K-steps.
// Branch-free body: out-of-range N is handled by clamping the W row pointer
// (junk only feeds D columns that are never stored), K tail by clamped
// indices + float masks. EXEC is all-1s at every WMMA.
template <bool A_FEATMAJOR>
__global__ __launch_bounds__(32)
void wmma_gemm_f32(const float* __restrict__ A, int lda,
                   const float* __restrict__ W, int K, int N, int kChunk,
                   float* __restrict__ out) {
    const int m0   = blockIdx.x * 16;
    const int n0   = blockIdx.y * 16;
    const int lane = threadIdx.x;      // 0..31
    const int half = lane >> 4;        // 0 | 1
    const int l15  = lane & 15;
    const int koff = half * 2;         // K offset of this lane's pair

    const int mA  = m0 + l15;          // A row for this lane (both halves)
    const int nB  = n0 + l15;          // B column for this lane
    const bool nOK = (nB < N);
    const int nBc = nOK ? nB : (N - 1);  // clamped safe W row

    const float* __restrict__ Wp = W + (size_t)nBc * K;
    const float* __restrict__ Ap = A_FEATMAJOR ? (A + mA)
                                               : (A + (size_t)mA * lda);

    const int kBegin = blockIdx.z * kChunk;
    const int kEnd   = (kBegin + kChunk < K) ? (kBegin + kChunk) : K;
    const int kMain  = kBegin + ((kEnd - kBegin) & ~3);

    __builtin_prefetch(Wp + kBegin, 0, 0);
    if (!A_FEATMAJOR) __builtin_prefetch(Ap + kBegin, 0, 0);

    v8f c = {0.f, 0.f, 0.f, 0.f, 0.f, 0.f, 0.f, 0.f};

    #pragma unroll 4
    for (int k0 = kBegin; k0 < kMain; k0 += 4) {
        const int k = k0 + koff;
        v2f a, b;
        if (A_FEATMAJOR) {
            a.x = Ap[(size_t)(k    ) * lda];
            a.y = Ap[(size_t)(k + 1) * lda];
        } else {
            const float2 av = *(const float2*)(Ap + k);
            a.x = av.x; a.y = av.y;
        }
        const float2 bv = *(const float2*)(Wp + k);
        b.x = bv.x; b.y = bv.y;
        c = __builtin_amdgcn_wmma_f32_16x16x4_f32(false, a, false, b,
                                                  (short)0, c, false, false);
    }

    if (kMain < kEnd) {  // 1..3 leftover K: clamped + masked, branch-free
        const int k  = kMain + koff;
        const int kx = (k     < kEnd) ? k       : (kEnd - 1);
        const int ky = (k + 1 < kEnd) ? (k + 1) : (kEnd - 1);
        const float mx = (k     < kEnd) ? 1.f : 0.f;
        const float my = (k + 1 < kEnd) ? 1.f : 0.f;
        v2f a, b;
        a.x = (A_FEATMAJOR ? Ap[(size_t)kx * lda] : Ap[kx]) * mx;
        a.y = (A_FEATMAJOR ? Ap[(size_t)ky * lda] : Ap[ky]) * my;
        b.x = Wp[kx] * mx;
        b.y = Wp[ky] * my;
        c = __builtin_amdgcn_wmma_f32_16x16x4_f32(false, a, false, b,
                                                  (short)0, c, false, false);
    }

    // D layout: VGPR i -> M = m0 + half*8 + i, N = n0 + l15.
    if (nOK) {
        const int mBase = m0 + half * 8;
        #pragma unroll
        for (int i = 0; i < 8; ++i)
            atomicAdd(&out[(size_t)nB * BATCH + mBase + i], c[i]);
    }
}

// ---------------------------------------------------------------------------
// Concat [h3(6), d3(6)] -> opre = f @ fW.T  (fb added in BN kernel).
__global__ void concat_fw(const float* __restrict__ h3, const float* __restrict__ d3,
                          const float* __restrict__ fW, float* __restrict__ opre) {
    const int b = threadIdx.x;
    float f[12];
    #pragma unroll
    for (int j = 0; j < 6; ++j) { f[j] = h3[j * BATCH + b]; f[6 + j] = d3[j * BATCH + b]; }
    #pragma unroll
    for (int i = 0; i < 6; ++i) {
        float s = 0.f;
        #pragma unroll
        for (int j = 0; j < 12; ++j) s += fW[i * 12 + j] * f[j];
        opre[i * BATCH + b] = s;
    }
}

// out[b] = tanh(obn[:,b] . aW + ab) * oW + ob
__global__ void final_k(const float* __restrict__ obn, const float* __restrict__ aW,
                        const float* __restrict__ ab,  const float* __restrict__ oW,
                        const float* __restrict__ ob,  float* __restrict__ out) {
    const int b = threadIdx.x;
    float s = ab[0];
    #pragma unroll
    for (int j = 0; j < 6; ++j) s += obn[j * BATCH + b] * aW[j];
    out[b] = tanhf(s) * oW[0] + ob[0];
}

// ---------------------------------------------------------------------------
extern "C" void kernel_launch(void* const* d_in, const int* in_sizes, int n_in,
                              void* d_out, int out_size, void* d_ws, size_t ws_size,
                              hipStream_t stream) {
    const float* x        = (const float*)d_in[0];
    const int*   sl1_rows = (const int*)d_in[1];
    const int*   sl1_cols = (const int*)d_in[2];
    const float* sl1_w    = (const float*)d_in[3];
    const float* sl1_b    = (const float*)d_in[4];
    const float* bn1_g    = (const float*)d_in[5];
    const float* bn1_b    = (const float*)d_in[6];
    const int*   sl2_rows = (const int*)d_in[7];
    const int*   sl2_cols = (const int*)d_in[8];
    const float* sl2_w    = (const float*)d_in[9];
    const float* sl2_b    = (const float*)d_in[10];
    const float* bn2_g    = (const float*)d_in[11];
    const float* bn2_b    = (const float*)d_in[12];
    const int*   sl3_rows = (const int*)d_in[13];
    const int*   sl3_cols = (const int*)d_in[14];
    const float* sl3_w    = (const float*)d_in[15];
    const float* sl3_b    = (const float*)d_in[16];
    const float* bn3_g    = (const float*)d_in[17];
    const float* bn3_b    = (const float*)d_in[18];
    const float* dW1      = (const float*)d_in[19];
    const float* db1      = (const float*)d_in[20];
    const float* dg1      = (const float*)d_in[21];
    const float* dbb1     = (const float*)d_in[22];
    const float* dW2      = (const float*)d_in[23];
    const float* db2      = (const float*)d_in[24];
    const float* dg2      = (const float*)d_in[25];
    const float* dbb2     = (const float*)d_in[26];
    const float* dW3      = (const float*)d_in[27];
    const float* db3      = (const float*)d_in[28];
    const float* dg3      = (const float*)d_in[29];
    const float* dbb3     = (const float*)d_in[30];
    const float* fW       = (const float*)d_in[31];
    const float* fb       = (const float*)d_in[32];
    const float* fg       = (const float*)d_in[33];
    const float* fbb      = (const float*)d_in[34];
    const float* aW       = (const float*)d_in[35];
    const float* ab       = (const float*)d_in[36];
    const float* oW       = (const float*)d_in[37];
    const float* ob       = (const float*)d_in[38];

    const int nE1 = in_sizes[3];   // 180000
    const int nE2 = in_sizes[9];   // 144000
    const int nE3 = in_sizes[15];  // 18000
    const int D1 = 12000, D2 = 3000, D3 = 6;

    // Workspace carve-up (floats). All accumulator ("pre") buffers are
    // contiguous so a single zero_f covers sparse scatter + GEMM K-split.
    float* ws = (float*)d_ws;
    size_t off = 0;
    float* xT    = ws + off; off += (size_t)GENE * BATCH;   // 6000x256
    float* h1pre = ws + off; off += (size_t)D1 * BATCH;     // | contiguous
    float* h2pre = ws + off; off += (size_t)D2 * BATCH;     // | zeroed
    float* h3pre = ws + off; off += (size_t)D3 * BATCH;     // | region
    float* d1pre = ws + off; off += (size_t)100 * BATCH;    // |
    float* d2pre = ws + off; off += (size_t)50 * BATCH;     // |
    float* d3pre = ws + off; off += (size_t)6 * BATCH;      // |
    float* h1    = ws + off; off += (size_t)D1 * BATCH;
    float* h2    = ws + off; off += (size_t)D2 * BATCH;
    float* h3    = ws + off; off += (size_t)D3 * BATCH;
    float* d1    = ws + off; off += (size_t)100 * BATCH;
    float* d2    = ws + off; off += (size_t)50 * BATCH;
    float* d3    = ws + off; off += (size_t)6 * BATCH;
    float* opre  = ws + off; off += (size_t)6 * BATCH;
    float* obn   = ws + off; off += (size_t)6 * BATCH;

    // Zero accumulators (every call; harness doesn't re-poison).
    const int nz = (D1 + D2 + D3 + 100 + 50 + 6) * BATCH;
    zero_f<<<(nz + 255) / 256, 256, 0, stream>>>(h1pre, nz);

    // ---- sparseGO branch ------------------------------------------------
    transpose_gene<<<GENE, BATCH, 0, stream>>>(x, xT);
    sparse_scatter<<<nE1, BATCH, 0, stream>>>(xT, sl1_rows, sl1_cols, sl1_w, h1pre);
    bias_tanh_bn<<<D1, BATCH, 0, stream>>>(h1pre, sl1_b, bn1_g, bn1_b, h1);
    sparse_scatter<<<nE2, BATCH, 0, stream>>>(h1, sl2_rows, sl2_cols, sl2_w, h2pre);
    bias_tanh_bn<<<D2, BATCH, 0, stream>>>(h2pre, sl2_b, bn2_g, bn2_b, h2);
    sparse_scatter<<<nE3, BATCH, 0, stream>>>(h2, sl3_rows, sl3_cols, sl3_w, h3pre);
    bias_tanh_bn<<<D3, BATCH, 0, stream>>>(h3pre, sl3_b, bn3_g, bn3_b, h3);

    // ---- drug MLP branch (f32 WMMA GEMMs) -------------------------------
    // d1pre[100,256] += drug(256x2048) @ dW1.T ; drug = x[:, 6000:], lda=XLD
    // K=2048 split into 8 chunks of 256 for occupancy (16 x 7 x 8 waves).
    wmma_gemm_f32<false><<<dim3(BATCH / 16, (100 + 15) / 16, 8), 32, 0, stream>>>(
        x + GENE, XLD, dW1, 2048, 100, 256, d1pre);
    bias_tanh_bn<<<100, BATCH, 0, stream>>>(d1pre, db1, dg1, dbb1, d1);

    wmma_gemm_f32<true><<<dim3(BATCH / 16, (50 + 15) / 16, 1), 32, 0, stream>>>(
        d1, BATCH, dW2, 100, 50, 100, d2pre);
    bias_tanh_bn<<<50, BATCH, 0, stream>>>(d2pre, db2, dg2, dbb2, d2);

    wmma_gemm_f32<true><<<dim3(BATCH / 16, 1, 1), 32, 0, stream>>>(
        d2, BATCH, dW3, 50, 6, 50, d3pre);
    bias_tanh_bn<<<6, BATCH, 0, stream>>>(d3pre, db3, dg3, dbb3, d3);

    // ---- head -----------------------------------------------------------
    concat_fw<<<1, BATCH, 0, stream>>>(h3, d3, fW, opre);
    bias_tanh_bn<<<6, BATCH, 0, stream>>>(opre, fb, fg, fbb, obn);
    final_k<<<1, BATCH, 0, stream>>>(obn, aW, ab, oW, ob, (float*)d_out);
}